// super_fusion_40570261078388
// MI455X (gfx1250) — compile-verified
//
#include <hip/hip_runtime.h>

typedef __attribute__((ext_vector_type(2))) float v2f;
typedef __attribute__((ext_vector_type(8))) float v8f;

// Problem geometry (fixed by the reference): b=4, c=32, t=64, h=w=64
#define NB 4
#define NC 32
#define NT 64
#define HW 4096                      // h*w
#define CHSTRIDE (NT * HW)           // stride between channels = 262144 elems
#define BSTRIDE  (NC * NT * HW)      // stride between batches  = 8388608 elems
#define OUT_ELEMS ((size_t)NB * NC * NT * HW)   // 33,554,432

// --------------------------------------------------------------------------
// K1: one block per (b,t) slice. 512 threads = 16 waves (wave32).
// Each wave computes comp for 16 pixels per iteration via chained
// v_wmma_f32_16x16x4_f32 over the 32-channel contraction, stores comp to
// LDS, block-reduces the L1 norm, and writes weight = 4096*comp/(2*l1)
// directly into the weight region of d_out.
// --------------------------------------------------------------------------
__global__ __launch_bounds__(512) void comp_weight_kernel(
    const float* __restrict__ app,   // [b,c,t,h,w]
    const float* __restrict__ W,     // [32]
    const float* __restrict__ b_lin, // [1]
    float* __restrict__ weight_out)  // [b,t,h,w]
{
    __shared__ float lds_comp[HW];   // 16 KB
    __shared__ float red[512];

    const int slice = blockIdx.x;        // 0..255  -> (b,t)
    const int bb = slice >> 6;
    const int tt = slice & 63;
    const float* base = app + (size_t)bb * BSTRIDE + (size_t)tt * HW;

    const int tid  = threadIdx.x;
    const int wave = tid >> 5;
    const int lane = tid & 31;
    const int hi   = (lane >> 4) & 1;    // which half of the wave
    const int pix  = lane & 15;          // pixel within 16-wide group
    const float bias = b_lin[0];

    // A-matrix operands: W replicated across rows. Lane half selects k-pair.
    float a0[8], a1[8];
#pragma unroll
    for (int j = 0; j < 8; ++j) {
        a0[j] = W[4 * j + 2 * hi];
        a1[j] = W[4 * j + 1 + 2 * hi];
    }

    float partial = 0.0f;

    for (int it = 0; it < 16; ++it) {
        const int pixBase = it * 256 + wave * 16;       // 16 waves * 16 px
        // B-matrix loads: lanes 0-15 fetch channels 4j,4j+1 for 16
        // consecutive pixels (one 64B line each); lanes 16-31 fetch
        // channels 4j+2,4j+3. Every byte fetched is consumed.
        const float* Xp = base + pixBase + pix + (size_t)(2 * hi) * CHSTRIDE;

        v8f acc = {};
#pragma unroll
        for (int j = 0; j < 8; ++j) {
            float b0v = Xp[(size_t)(4 * j) * CHSTRIDE];
            float b1v = Xp[(size_t)(4 * j + 1) * CHSTRIDE];
            v2f Bv = {b0v, b1v};
            v2f Av = {a0[j], a1[j]};
            // D[m][n] = sum_k W[4j+k] * X[4j+k][pixel n]  (same for all m)
            acc = __builtin_amdgcn_wmma_f32_16x16x4_f32(
                false, Av, false, Bv, (short)0, acc, false, false);
        }

        // acc[0] holds comp_lin for pixel (lane & 15), duplicated in halves
        float lin  = acc[0] + bias;
        // sigmoid via fast hardware reciprocal (v_rcp_f32, ~1 ulp) instead of
        // the 11-op IEEE division expansion; result > 0, so |comp| == comp.
        float comp = __builtin_amdgcn_rcpf(1.0f + __expf(-lin));
        partial += comp;
        if (hi == 0) lds_comp[pixBase + pix] = comp;
    }

    // Deterministic block reduction. Each pixel counted twice (two halves),
    // so red[0] == 2 * l1.
    red[tid] = partial;
    __syncthreads();
    for (int s = 256; s > 0; s >>= 1) {
        if (tid < s) red[tid] += red[tid + s];
        __syncthreads();
    }
    // weight = (h*w) * comp / (2 * l1) = 4096 * comp / red[0]
    const float scale = 4096.0f / red[0];

    float4* wo4 = (float4*)(weight_out + (size_t)slice * HW);
    const float4* lc4 = (const float4*)lds_comp;
    for (int i = tid; i < HW / 4; i += 512) {
        float4 v = lc4[i];
        float4 o = {v.x * scale, v.y * scale, v.z * scale, v.w * scale};
        wo4[i] = o;
    }
}

// --------------------------------------------------------------------------
// K2: out[b,c,t,h,w] = motion[b,c,t,h,w] * weight[b,t,h,w]
// Pure HBM stream with float4 accesses; weight region (4 MB) is L2-resident.
// One block handles 1024 elements of one (b,c,t) plane.
// --------------------------------------------------------------------------
__global__ __launch_bounds__(256) void scale_motion_kernel(
    const float* __restrict__ motion,
    const float* __restrict__ weight,
    float* __restrict__ out)
{
    const int chunk = blockIdx.x;            // 32768 chunks
    const int plane = chunk >> 2;            // (b*32 + c)*64 + t
    const int off   = ((chunk & 3) << 10) + (threadIdx.x << 2);
    const int bb = plane >> 11;              // plane / (32*64)
    const int tt = plane & 63;

    const size_t mIdx = (size_t)plane * HW + off;
    const size_t wIdx = ((size_t)(bb * NT + tt)) * HW + off;

    float4 m = *(const float4*)(motion + mIdx);
    float4 w = *(const float4*)(weight + wIdx);
    float4 o = {m.x * w.x, m.y * w.y, m.z * w.z, m.w * w.w};
    *(float4*)(out + mIdx) = o;
}

extern "C" void kernel_launch(void* const* d_in, const int* in_sizes, int n_in,
                              void* d_out, int out_size, void* d_ws, size_t ws_size,
                              hipStream_t stream) {
    (void)in_sizes; (void)n_in; (void)out_size; (void)d_ws; (void)ws_size;
    const float* motion = (const float*)d_in[0];   // [4,32,64,64,64]
    const float* app    = (const float*)d_in[1];   // [4,32,64,64,64]
    const float* W      = (const float*)d_in[2];   // [32]
    const float* b_lin  = (const float*)d_in[3];   // [1]

    float* out    = (float*)d_out;                 // first output, 33.5M floats
    float* weight = out + OUT_ELEMS;               // second output, 1.05M floats

    // K1: 256 slices, 512 threads (16 wave32 waves) each
    comp_weight_kernel<<<NB * NT, 512, 0, stream>>>(app, W, b_lin, weight);
    // K2: 8192 planes * 4 chunks, 256 threads * float4
    scale_motion_kernel<<<NB * NC * NT * 4, 256, 0, stream>>>(motion, weight, out);
}